// Graph_Conv_Block_25950192402911
// MI455X (gfx1250) — compile-verified
//
#include <hip/hip_runtime.h>
#include <hip/hip_bf16.h>
#include <stdint.h>

#define D_FEAT 128
#define STRIDE2 144         // float2 stride per K-pair row: 2*144 mod 64 == 32 dwords
                            // -> the two lane-halves hit disjoint 32-bank halves (conflict-free b64)
#define LEAKY_SLOPE 0.01f
#define BN_EPS 1e-5f

typedef __attribute__((ext_vector_type(2))) float v2f;
typedef __attribute__((ext_vector_type(8))) float v8f;

// ---------- init ----------
__global__ void zero_f32_kernel(float* __restrict__ p, long n) {
    long i = (long)blockIdx.x * blockDim.x + threadIdx.x;
    if (i < n) p[i] = 0.0f;
}

__global__ void init_deg_kernel(float* __restrict__ deg, int N) {
    int i = blockIdx.x * blockDim.x + threadIdx.x;
    if (i < N) deg[i] = 1.0f;   // self-loop
}

// ---------- degree / dinv ----------
__global__ void degree_kernel(const long long* __restrict__ dst,
                              float* __restrict__ deg, int E) {
    int e = blockIdx.x * blockDim.x + threadIdx.x;
    if (e < E) unsafeAtomicAdd(&deg[(int)dst[e]], 1.0f);
}

__global__ void dinv_kernel(const float* __restrict__ deg,
                            float* __restrict__ dinv, int N) {
    int i = blockIdx.x * blockDim.x + threadIdx.x;
    if (i < N) dinv[i] = rsqrtf(deg[i]);
}

// ---------- GEMM h = x @ W via V_WMMA_F32_16X16X4_F32 ----------
// Workgroup = 256 threads = 8 waves; each wave computes 16 rows x 128 cols.
// A 16x4 fp32: lanes0-15 carry K={k0,k0+1}, lanes16-31 K={k0+2,k0+3}  (one 8B load).
// B 4x16:      VGPR0 = K0/K2 rows, VGPR1 = K1/K3 — W staged in LDS as float2
//              pairs {W[k][c],W[k+1][c]} so each B operand is one ds_load_b64.
// C/D 16x16:   VGPR r -> M=r (lanes0-15), M=r+8 (lanes16-31), N = lane%16.
__global__ __launch_bounds__(256)
void gemm_wmma_kernel(const float* __restrict__ x,
                      const float* __restrict__ W,
                      float* __restrict__ h, int N) {
    __shared__ v2f Wlds[64 * STRIDE2];     // 64 K-pair rows x 128 cols (padded), 73728 B
    const int tid = threadIdx.x;

    // Cooperative stage: pair rows k=2*k2 and 2*k2+1 per column.
    for (int idx = tid; idx < 64 * D_FEAT; idx += 256) {
        int k2 = idx >> 7, c = idx & 127;
        v2f w;
        w.x = W[(2 * k2) * D_FEAT + c];
        w.y = W[(2 * k2 + 1) * D_FEAT + c];
        Wlds[k2 * STRIDE2 + c] = w;
    }
    __syncthreads();

    const int wave    = tid >> 5;
    const int lane    = tid & 31;
    const int lane_lo = lane & 15;   // M (A) / N (B,C)
    const int lane_hi = lane >> 4;   // K-half select

    const long row_base = (long)blockIdx.x * 128 + wave * 16;
    long arow = row_base + lane_lo;
    if (arow >= N) arow = N - 1;     // clamp loads; stores guarded below
    const float* xrow = x + arow * D_FEAT;

    v8f acc[8];
#pragma unroll
    for (int n = 0; n < 8; ++n)
#pragma unroll
        for (int i = 0; i < 8; ++i) acc[n][i] = 0.0f;

    for (int k0 = 0; k0 < D_FEAT; k0 += 4) {
        const int kp = (k0 >> 1) + lane_hi;                  // K-pair row for this lane half
        v2f a = *(const v2f*)(xrow + k0 + 2 * lane_hi);      // one 8B global load
        const v2f* brow = &Wlds[kp * STRIDE2 + lane_lo];
#pragma unroll
        for (int n = 0; n < 8; ++n) {
            v2f b = brow[n * 16];                            // one ds_load_b64
            acc[n] = __builtin_amdgcn_wmma_f32_16x16x4_f32(
                /*neg_a=*/false, a, /*neg_b=*/false, b,
                /*c_mod=*/(short)0, acc[n],
                /*reuse_a=*/false, /*reuse_b=*/false);
        }
    }

    // Epilogue: fast unguarded path for full tiles (all but the last block).
    if (row_base + 16 <= (long)N) {
        float* pbase = h + (row_base + 8 * lane_hi) * D_FEAT + lane_lo;
#pragma unroll
        for (int n = 0; n < 8; ++n) {
            float* p = pbase + n * 16;
#pragma unroll
            for (int r = 0; r < 8; ++r)
                p[r * D_FEAT] = acc[n][r];   // imm-offset global_store_b32
        }
    } else {
#pragma unroll
        for (int n = 0; n < 8; ++n) {
            const int col = n * 16 + lane_lo;
#pragma unroll
            for (int r = 0; r < 8; ++r) {
                long row = row_base + r + 8 * lane_hi;
                if (row < N) h[row * D_FEAT + col] = acc[n][r];
            }
        }
    }
}

// ---------- edge scatter: agg[dst] += dinv[src]*dinv[dst] * h[src] ----------
// 2 edges per 256-thread block; one feature per thread (coalesced 512B/edge).
__global__ __launch_bounds__(256)
void scatter_kernel(const long long* __restrict__ src,
                    const long long* __restrict__ dst,
                    const float* __restrict__ dinv,
                    const float* __restrict__ h,
                    float* __restrict__ agg, int E) {
    int e = blockIdx.x * 2 + (threadIdx.x >> 7);
    int f = threadIdx.x & 127;
    if (e >= E) return;
    int s = (int)src[e];
    int d = (int)dst[e];
    float coef = dinv[s] * dinv[d];
    unsafeAtomicAdd(&agg[(long)d * D_FEAT + f], coef * h[(long)s * D_FEAT + f]);
}

// ---------- self-loop + bias + LeakyReLU + BN partial stats ----------
__global__ __launch_bounds__(128)
void post_kernel(const float* __restrict__ h,
                 const float* __restrict__ agg,
                 const float* __restrict__ dinv,
                 const float* __restrict__ bvec,
                 float* __restrict__ outpre,
                 float* __restrict__ sums,
                 float* __restrict__ sqs, int N) {
    const int f = threadIdx.x;
    const int base = blockIdx.x * 64;
    const float bv = bvec[f];
    float lsum = 0.0f, lsq = 0.0f;
    for (int r = 0; r < 64; ++r) {
        int node = base + r;            // uniform across block
        if (node >= N) break;
        float di = dinv[node];
        float invd = di * di;           // 1/deg
        long idx = (long)node * D_FEAT + f;
        float val = agg[idx] + h[idx] * invd + bv;
        val = (val >= 0.0f) ? val : LEAKY_SLOPE * val;
        outpre[idx] = val;
        lsum += val;
        lsq  += val * val;
    }
    unsafeAtomicAdd(&sums[f], lsum);
    unsafeAtomicAdd(&sqs[f], lsq);
}

// ---------- BN finalize (in place on d_out) ----------
__global__ void bn_kernel(float* __restrict__ out,
                          const float* __restrict__ sums,
                          const float* __restrict__ sqs,
                          const float* __restrict__ gamma,
                          const float* __restrict__ beta, int N) {
    const long total = (long)N * D_FEAT;
    const float invN = 1.0f / (float)N;
    for (long idx = (long)blockIdx.x * blockDim.x + threadIdx.x;
         idx < total; idx += (long)gridDim.x * blockDim.x) {
        int f = (int)(idx & 127);
        float mean = sums[f] * invN;
        float var  = sqs[f] * invN - mean * mean;   // biased var
        out[idx] = (out[idx] - mean) * rsqrtf(var + BN_EPS) * gamma[f] + beta[f];
    }
}

extern "C" void kernel_launch(void* const* d_in, const int* in_sizes, int n_in,
                              void* d_out, int out_size, void* d_ws, size_t ws_size,
                              hipStream_t stream) {
    const float*     x     = (const float*)d_in[0];
    const long long* edge  = (const long long*)d_in[1];   // int64 [2, E]
    const float*     W     = (const float*)d_in[2];
    const float*     b     = (const float*)d_in[3];
    const float*     gamma = (const float*)d_in[4];
    const float*     beta  = (const float*)d_in[5];

    const int N = in_sizes[0] / D_FEAT;
    const int E = in_sizes[1] / 2;
    const long long* srcI = edge;
    const long long* dstI = edge + E;

    // Workspace layout (floats): h[N*128] | agg[N*128] | sums[128] | sqs[128] | deg[N] | dinv[N]
    float* ws   = (float*)d_ws;
    float* h    = ws;
    float* agg  = h   + (long)N * D_FEAT;
    float* sums = agg + (long)N * D_FEAT;
    float* sqs  = sums + D_FEAT;
    float* deg  = sqs  + D_FEAT;
    float* dinv = deg  + N;
    float* out  = (float*)d_out;

    // Re-initialize mutable scratch every call (graph replay safe).
    long zlen = (long)N * D_FEAT + 2 * D_FEAT;     // agg + sums + sqs contiguous
    zero_f32_kernel<<<(int)((zlen + 255) / 256), 256, 0, stream>>>(agg, zlen);
    init_deg_kernel<<<(N + 255) / 256, 256, 0, stream>>>(deg, N);

    degree_kernel<<<(E + 255) / 256, 256, 0, stream>>>(dstI, deg, E);
    dinv_kernel<<<(N + 255) / 256, 256, 0, stream>>>(deg, dinv, N);

    gemm_wmma_kernel<<<(N + 127) / 128, 256, 0, stream>>>(x, W, h, N);

    scatter_kernel<<<(E + 1) / 2, 256, 0, stream>>>(srcI, dstI, dinv, h, agg, E);

    post_kernel<<<(N + 63) / 64, 128, 0, stream>>>(h, agg, dinv, b, out, sums, sqs, N);

    bn_kernel<<<2048, 256, 0, stream>>>(out, sums, sqs, gamma, beta, N);
}